// Qwen3Attention_27951647162496
// MI455X (gfx1250) — compile-verified
//
#include <hip/hip_runtime.h>
#include <hip/hip_bf16.h>

typedef __attribute__((ext_vector_type(2))) float v2f;
typedef __attribute__((ext_vector_type(4))) float v4f;
typedef __attribute__((ext_vector_type(8))) float v8f;
typedef int v4i __attribute__((vector_size(4 * sizeof(int))));

typedef __attribute__((address_space(1))) v4i* gbl_v4i_p;   // global (AS1)
typedef __attribute__((address_space(3))) v4i* lds_v4i_p;   // LDS (AS3)

// D = A(16x4 f32) * B(4x16 f32) + C(16x16 f32), wave32 CDNA5 WMMA.
__device__ __forceinline__ v8f wmma_f32(v2f a, v2f b, v8f c) {
    return __builtin_amdgcn_wmma_f32_16x16x4_f32(
        /*neg_a=*/false, a, /*neg_b=*/false, b,
        /*c_mod=*/(short)0, c, /*reuse_a=*/false, /*reuse_b=*/false);
}

__device__ __forceinline__ v8f v8f_zero() {
    v8f z = {0.f, 0.f, 0.f, 0.f, 0.f, 0.f, 0.f, 0.f};
    return z;
}

// Async DMA: 16 bytes global -> LDS, tracked by ASYNCcnt (no VGPR staging).
__device__ __forceinline__ void async16(const float* g, float* l) {
    __builtin_amdgcn_global_load_async_to_lds_b128(
        (gbl_v4i_p)(g), (lds_v4i_p)(l), /*offset=*/0, /*cpol=*/0);
}

__device__ __forceinline__ void wait_async0() {
    asm volatile("s_wait_asynccnt 0x0" ::: "memory");
}

// ---------------------------------------------------------------------------
// Generic C[M x N] = A[M x K] * W[N x K]^T, fp32 WMMA 16x16x4.
// Macro tile 128(M) x 128(N), 256 threads = 8 waves laid out 4(M) x 2(N);
// each wave owns a 32x64 tile = 2x4 WMMA accumulators.
// Double-buffered LDS filled by GLOBAL_LOAD_ASYNC_TO_LDS_B128; one barrier
// per K stage. Rows padded to stride 20 floats (conflict-free, 16B aligned).
// ---------------------------------------------------------------------------
__global__ __launch_bounds__(256)
void gemm_xwT(const float* __restrict__ A, const float* __restrict__ W,
              float* __restrict__ C, int K, int sA, int sC)
{
    __shared__ float As[2][128 * 20];
    __shared__ float Ws[2][128 * 20];

    const int tid  = threadIdx.x;
    const int lane = tid & 31;
    const int wave = tid >> 5;
    const int hi   = lane >> 4;   // K half select {0,1} -> k pair {0,1} vs {2,3}
    const int ln   = lane & 15;   // M row (A frag) / N col (B frag, C/D)
    const int wm   = wave & 3;    // wave M index 0..3
    const int wn   = wave >> 2;   // wave N index 0..1
    const long m0  = (long)blockIdx.x * 128;
    const long n0  = (long)blockIdx.y * 128;

    // Per-thread staging slots: 2 x b128 each for the A tile and the W tile.
    const int id0  = tid * 2;
    const int row0 = id0 >> 2,       c40 = (id0 & 3) * 4;       // 0,4,8,12
    const int row1 = (id0 + 1) >> 2, c41 = ((id0 + 1) & 3) * 4;

    const float* a0p = A + (m0 + row0) * (long)sA + c40;
    const float* a1p = A + (m0 + row1) * (long)sA + c41;
    const float* w0p = W + (n0 + row0) * (long)K  + c40;
    const float* w1p = W + (n0 + row1) * (long)K  + c41;
    const int la0 = row0 * 20 + c40;
    const int la1 = row1 * 20 + c41;

    v8f acc[2][4];
#pragma unroll
    for (int i = 0; i < 2; ++i)
#pragma unroll
        for (int j = 0; j < 4; ++j) acc[i][j] = v8f_zero();

    // Prologue: DMA stage 0.
    async16(a0p, &As[0][la0]);
    async16(a1p, &As[0][la1]);
    async16(w0p, &Ws[0][la0]);
    async16(w1p, &Ws[0][la1]);

    int st = 0;
    for (int kk = 0; kk < K; kk += 16, st ^= 1) {
        wait_async0();        // this wave's DMA into buffer st is complete
        __syncthreads();      // ... and visible to all waves; st^1 is free

        if (kk + 16 < K) {    // DMA next slab into the other buffer
            const int nk = kk + 16;
            async16(a0p + nk, &As[st ^ 1][la0]);
            async16(a1p + nk, &As[st ^ 1][la1]);
            async16(w0p + nk, &Ws[st ^ 1][la0]);
            async16(w1p + nk, &Ws[st ^ 1][la1]);
        }

        const float* as = As[st];
        const float* ws = Ws[st];
#pragma unroll
        for (int c = 0; c < 4; ++c) {
            const int ko = c * 4 + 2 * hi;
            v2f a0 = *(const v2f*)(&as[(wm * 32 +      ln) * 20 + ko]);
            v2f a1 = *(const v2f*)(&as[(wm * 32 + 16 + ln) * 20 + ko]);
#pragma unroll
            for (int nt = 0; nt < 4; ++nt) {
                v2f bw = *(const v2f*)(&ws[(wn * 64 + nt * 16 + ln) * 20 + ko]);
                acc[0][nt] = wmma_f32(a0, bw, acc[0][nt]);
                acc[1][nt] = wmma_f32(a1, bw, acc[1][nt]);
            }
        }
    }

#pragma unroll
    for (int mt = 0; mt < 2; ++mt)
#pragma unroll
        for (int nt = 0; nt < 4; ++nt)
#pragma unroll
            for (int r = 0; r < 8; ++r) {
                const long row = m0 + wm * 32 + mt * 16 + r + 8 * hi;
                const long col = n0 + wn * 64 + nt * 16 + ln;
                C[row * (long)sC + col] = acc[mt][nt][r];
            }
}

// ---------------------------------------------------------------------------
// Per-head RMSNorm + RoPE, in place on qkv[B*S, 6144]. One wave per head
// (32 q heads + 8 k heads); lane l owns d = {l, l+32, l+64, l+96}.
// ---------------------------------------------------------------------------
__global__ __launch_bounds__(256)
void rmsnorm_rope(float* __restrict__ qkv, const int* __restrict__ positions,
                  const float* __restrict__ qw, const float* __restrict__ kw)
{
    const int t    = blockIdx.x;           // token in [0, B*S)
    const int lane = threadIdx.x & 31;
    const int wave = threadIdx.x >> 5;
    const float pos = (float)positions[t];
    const float lt  = logf(1.0e6f);        // log(THETA)
    float* base = qkv + (long)t * 6144;

    for (int hd = wave; hd < 40; hd += 8) {
        float* p;
        const float* w;
        if (hd < 32) { p = base + hd * 128;               w = qw; }
        else         { p = base + 4096 + (hd - 32) * 128; w = kw; }

        float x0 = p[lane], x1 = p[lane + 32], x2 = p[lane + 64], x3 = p[lane + 96];
        float ss = x0 * x0 + x1 * x1 + x2 * x2 + x3 * x3;
#pragma unroll
        for (int off = 16; off > 0; off >>= 1) ss += __shfl_xor(ss, off, 32);
        const float inv = rsqrtf(ss * (1.0f / 128.0f) + 1e-6f);

        const float y0 = x0 * inv * w[lane];
        const float y1 = x1 * inv * w[lane + 32];
        const float y2 = x2 * inv * w[lane + 64];
        const float y3 = x3 * inv * w[lane + 96];

        const float fa = pos * expf(-(2.0f * (float)lane        / 128.0f) * lt);
        const float fb = pos * expf(-(2.0f * (float)(lane + 32) / 128.0f) * lt);
        const float ca = cosf(fa), sa = sinf(fa);
        const float cb = cosf(fb), sb = sinf(fb);

        p[lane]      = y0 * ca - y2 * sa;   // d = l        (pairs with d+64)
        p[lane + 64] = y2 * ca + y0 * sa;
        p[lane + 32] = y1 * cb - y3 * sb;   // d = l+32
        p[lane + 96] = y3 * cb + y1 * sb;
    }
}

// ---------------------------------------------------------------------------
// Causal GQA flash attention, fp32 WMMA. One wave per 16-row Q tile; online
// softmax; P (C-layout) -> A-layout transpose through private padded LDS tile.
// Output overwrites the q slot of the qkv buffer. K/V fragments stream from
// global (2 MB per (b, kv-head) -> L2 resident, reused 4x by GQA + 16 blocks).
// ---------------------------------------------------------------------------
__global__ __launch_bounds__(256)
void flash_attn(float* __restrict__ qkv, int S_)
{
    __shared__ float ptile[8][16 * 20];

    const int lane = threadIdx.x & 31;
    const int wave = threadIdx.x >> 5;
    const int hi   = lane >> 4;
    const int ln   = lane & 15;
    const int h    = blockIdx.y;
    const int b    = blockIdx.z;
    const int kvh  = h >> 2;                     // rep = H/KV = 4
    const int q0   = blockIdx.x * 128 + wave * 16;
    const long ts  = 6144;                       // token stride (floats)

    const float* qptr  = qkv + ((long)b * S_ + q0 + ln) * ts + h * 128;
    const float* kbase = qkv + ((long)b * S_) * ts + 4096 + kvh * 128;
    const float* vbase = qkv + ((long)b * S_) * ts + 5120 + kvh * 128;
    const float scale = 0.08838834764831845f;    // 1/sqrt(128)

    // Preload scaled Q row fragments (A layout): 32 chunks of float2 per lane.
    v2f qa[32];
#pragma unroll
    for (int c = 0; c < 32; ++c) {
        v2f q = *(const v2f*)(qptr + c * 4 + 2 * hi);
        qa[c].x = q.x * scale;
        qa[c].y = q.y * scale;
    }

    v8f o[8];
#pragma unroll
    for (int t = 0; t < 8; ++t) o[t] = v8f_zero();
    float rm[8], rl[8];
#pragma unroll
    for (int r = 0; r < 8; ++r) { rm[r] = -3.0e38f; rl[r] = 0.0f; }

    const int ntile = q0 / 16 + 1;               // full tiles + diagonal tile

    for (int kt = 0; kt < ntile; ++kt) {
        const int k0 = kt * 16;

        // S_tile = Q * K^T  (B frag lane = key position ln of this tile)
        const float* krow = kbase + (long)(k0 + ln) * ts;
        v8f s = v8f_zero();
#pragma unroll
        for (int c = 0; c < 32; ++c) {
            v2f kb = *(const v2f*)(krow + c * 4 + 2 * hi);
            s = wmma_f32(qa[c], kb, s);
        }

        if (kt == ntile - 1) {                   // causal mask (diagonal tile)
#pragma unroll
            for (int r = 0; r < 8; ++r) {
                const int row = q0 + r + 8 * hi;
                if (k0 + ln > row) s[r] = -3.0e38f;
            }
        }

        // Online softmax: row stats replicated across each 16-lane half.
#pragma unroll
        for (int r = 0; r < 8; ++r) {
            float mx = s[r];
#pragma unroll
            for (int off = 8; off > 0; off >>= 1) mx = fmaxf(mx, __shfl_xor(mx, off, 32));
            const float mnew  = fmaxf(rm[r], mx);
            const float alpha = __expf(rm[r] - mnew);
            rm[r] = mnew;
            const float pv = __expf(s[r] - mnew);
            float sum = pv;
#pragma unroll
            for (int off = 8; off > 0; off >>= 1) sum += __shfl_xor(sum, off, 32);
            rl[r] = rl[r] * alpha + sum;
#pragma unroll
            for (int t = 0; t < 8; ++t) o[t][r] *= alpha;
            ptile[wave][(r + 8 * hi) * 20 + ln] = pv;   // C-layout -> LDS
        }

        // O += P * V   (P re-read from LDS in A layout)
        const float* vk = vbase + (long)k0 * ts;
#pragma unroll
        for (int c = 0; c < 4; ++c) {
            const int ko = c * 4 + 2 * hi;
            v2f pa = *(const v2f*)(&ptile[wave][ln * 20 + ko]);
#pragma unroll
            for (int t = 0; t < 8; ++t) {
                const int d = t * 16 + ln;
                v2f vb;
                vb.x = vk[(long)ko * ts + d];
                vb.y = vk[(long)(ko + 1) * ts + d];
                o[t] = wmma_f32(pa, vb, o[t]);
            }
        }
    }

    // Normalize and write into the q slot (safe: Q already consumed).
    float* optr = qkv + ((long)b * S_ + q0) * ts + h * 128;
#pragma unroll
    for (int r = 0; r < 8; ++r) {
        const float inv = 1.0f / rl[r];
        const int row = r + 8 * hi;
#pragma unroll
        for (int t = 0; t < 8; ++t)
            optr[(long)row * ts + t * 16 + ln] = o[t][r] * inv;
    }
}

// ---------------------------------------------------------------------------
extern "C" void kernel_launch(void* const* d_in, const int* in_sizes, int n_in,
                              void* d_out, int out_size, void* d_ws, size_t ws_size,
                              hipStream_t stream)
{
    const float* x         = (const float*)d_in[0];
    const int*   positions = (const int*)  d_in[1];
    const float* w_qkv     = (const float*)d_in[2];
    const float* q_norm_w  = (const float*)d_in[3];
    const float* k_norm_w  = (const float*)d_in[4];
    const float* w_o       = (const float*)d_in[5];
    float* out = (float*)d_out;
    float* qkv = (float*)d_ws;       // [B*S, 6144] fp32 = ~100.7 MB

    const dim3 blk(256);

    // 1) qkv = x @ w_qkv^T : M=4096, N=6144, K=4096
    gemm_xwT<<<dim3(4096 / 128, 6144 / 128), blk, 0, stream>>>(
        x, w_qkv, qkv, 4096, 4096, 6144);

    // 2) RMSNorm + RoPE in place (q + k heads)
    rmsnorm_rope<<<dim3(2 * 2048), blk, 0, stream>>>(
        qkv, positions, q_norm_w, k_norm_w);

    // 3) causal GQA flash attention (writes O into q slot of qkv)
    flash_attn<<<dim3(2048 / 128, 32, 2), blk, 0, stream>>>(qkv, 2048);

    // 4) out = O @ w_o^T : M=4096, N=4096, K=4096, A row stride 6144
    gemm_xwT<<<dim3(4096 / 128, 4096 / 128), blk, 0, stream>>>(
        qkv, w_o, out, 4096, 6144, 4096);
}